// MessageLoss_42339787604589
// MI455X (gfx1250) — compile-verified
//
#include <hip/hip_runtime.h>

// ---- vector types --------------------------------------------------------
typedef __attribute__((ext_vector_type(2))) float v2f;
typedef __attribute__((ext_vector_type(8))) float v8f;
// 4-byte-aligned wide loads (rows of 7 floats are only DWORD aligned;
// gfx1250 DWORD alignment mode allows b128/b64 at 4B alignment)
typedef __attribute__((ext_vector_type(4), aligned(4))) float f4u;
typedef __attribute__((ext_vector_type(2), aligned(4))) float f2u;

// ---- 32-lane wave sum via V_WMMA_F32_16X16X4_F32 -------------------------
// A (16x4 f32): lane L puts its value v in VGPR0 (K=0 for L<16, K=2 for L>=16),
// VGPR1 = 0.  B (4x16) = all ones.  D[m][n] = v[m] + v[m+16] for all n.
// C/D layout: VGPR r, lanes 0-15 -> M=r, lanes 16-31 -> M=r+8.  Summing the
// 8 D VGPRs per lane gives the lower/upper half-wave totals; one xor-16
// shuffle yields the full 32-lane sum in every lane.  Full f32 precision.
__device__ __forceinline__ float wave_sum32(float v) {
  v2f a; a[0] = v;    a[1] = 0.0f;
  v2f b; b[0] = 1.0f; b[1] = 1.0f;
  v8f c = {};
  c = __builtin_amdgcn_wmma_f32_16x16x4_f32(false, a, false, b,
                                            (short)0, c, false, false);
  float s = c[0] + c[1] + c[2] + c[3] + c[4] + c[5] + c[6] + c[7];
  s += __shfl_xor(s, 16, 32);
  return s;
}

// ---- workspace layout (floats) -------------------------------------------
// ws[0] = sum (labeled_doc - target1)^2
// ws[1] = sum (unlabeled_doc - target3)^2
// ws[2] = sum nll * mask
// ws[3+k], k=0..22 : [correct_total(7) | predict_dict(7) | correct_dict(7) | correct | count]

__global__ void zero_ws_kernel(float* __restrict__ ws) {
  if (threadIdx.x < 26) ws[threadIdx.x] = 0.0f;
}

// ---- kernel 1: both doc MSE sums in one streaming pass -------------------
__global__ void doc_mse_kernel(const float* __restrict__ ld,
                               const float* __restrict__ t1,
                               const float* __restrict__ ud,
                               const float* __restrict__ t3,
                               int n4, int n, float* __restrict__ ws) {
  __shared__ float l1s[32];
  __shared__ float l3s[32];
  float p1 = 0.0f, p3 = 0.0f;
  const int stride = gridDim.x * blockDim.x;
  const int gid = blockIdx.x * blockDim.x + threadIdx.x;
  for (int i = gid; i < n4; i += stride) {
    f4u a = *(const f4u*)(ld + 4 * (size_t)i);
    f4u b = *(const f4u*)(t1 + 4 * (size_t)i);
    f4u c = *(const f4u*)(ud + 4 * (size_t)i);
    f4u d = *(const f4u*)(t3 + 4 * (size_t)i);
#pragma unroll
    for (int k = 0; k < 4; ++k) {
      float e1 = a[k] - b[k]; p1 += e1 * e1;
      float e3 = c[k] - d[k]; p3 += e3 * e3;
    }
  }
  if (gid == 0) {                       // scalar tail (n % 4)
    for (int i = n4 * 4; i < n; ++i) {
      float e1 = ld[i] - t1[i]; p1 += e1 * e1;
      float e3 = ud[i] - t3[i]; p3 += e3 * e3;
    }
  }
  const int lane = threadIdx.x & 31;
  const int wv   = threadIdx.x >> 5;
  float s1 = wave_sum32(p1);
  float s3 = wave_sum32(p3);
  if (lane == 0) { l1s[wv] = s1; l3s[wv] = s3; }
  __syncthreads();
  if (wv == 0) {
    const int nw = blockDim.x >> 5;
    float v1 = (lane < nw) ? l1s[lane] : 0.0f;
    float v3 = (lane < nw) ? l3s[lane] : 0.0f;
    float b1 = wave_sum32(v1);
    float b3 = wave_sum32(v3);
    if (lane == 0) {
      atomicAdd(ws + 0, b1);
      atomicAdd(ws + 1, b3);
    }
  }
}

// ---- kernel 2: masked cross-entropy + argmax statistics ------------------
#define NUM_CLASSES 7
#define IGNORE_IDX  8

__global__ void sent_loss_kernel(const float* __restrict__ logits,
                                 const int* __restrict__ tgt,
                                 int n, float* __restrict__ ws) {
  __shared__ unsigned cnt[23];   // ct[7] | pd[7] | cd[7] | correct | count
  __shared__ float    nlls[32];
  if (threadIdx.x < 23) cnt[threadIdx.x] = 0u;
  __syncthreads();

  float pn = 0.0f;
  const int stride = gridDim.x * blockDim.x;
  for (int i = blockIdx.x * blockDim.x + threadIdx.x; i < n; i += stride) {
    const float* row = logits + (size_t)7 * (size_t)i;
    __builtin_prefetch(row + (size_t)7 * (size_t)stride, 0, 0); // global_prefetch_b8
    f4u v0 = *(const f4u*)row;            // x0..x3 (4B-aligned b128)
    f2u v1 = *(const f2u*)(row + 4);      // x4..x5
    float x0 = v0[0], x1 = v0[1], x2 = v0[2], x3 = v0[3];
    float x4 = v1[0], x5 = v1[1], x6 = row[6];
    int  t     = tgt[i];
    bool valid = (t != IGNORE_IDX);

    // argmax, first occurrence -> strict '>'
    float best = x0; int pred = 0;
    if (x1 > best) { best = x1; pred = 1; }
    if (x2 > best) { best = x2; pred = 2; }
    if (x3 > best) { best = x3; pred = 3; }
    if (x4 > best) { best = x4; pred = 4; }
    if (x5 > best) { best = x5; pred = 5; }
    if (x6 > best) { best = x6; pred = 6; }

    float se = __expf(x0 - best) + __expf(x1 - best) + __expf(x2 - best) +
               __expf(x3 - best) + __expf(x4 - best) + __expf(x5 - best) +
               __expf(x6 - best);
    float lse = best + __logf(se);

    int tt = valid ? t : 0;
    float xt = x0;
    xt = (tt == 1) ? x1 : xt;
    xt = (tt == 2) ? x2 : xt;
    xt = (tt == 3) ? x3 : xt;
    xt = (tt == 4) ? x4 : xt;
    xt = (tt == 5) ? x5 : xt;
    xt = (tt == 6) ? x6 : xt;

    if (valid) {
      pn += lse - xt;                         // nll for this row
      atomicAdd(&cnt[tt], 1u);                // correct_total[t]
      atomicAdd(&cnt[7 + pred], 1u);          // predict_dict[pred]
      if (pred == t) {
        atomicAdd(&cnt[14 + tt], 1u);         // correct_dict[t]
        atomicAdd(&cnt[21], 1u);              // correct
      }
      atomicAdd(&cnt[22], 1u);                // count
    }
  }

  const int lane = threadIdx.x & 31;
  const int wv   = threadIdx.x >> 5;
  float s = wave_sum32(pn);
  if (lane == 0) nlls[wv] = s;
  __syncthreads();
  if (wv == 0) {
    const int nw = blockDim.x >> 5;
    float v = (lane < nw) ? nlls[lane] : 0.0f;
    float tot = wave_sum32(v);
    if (lane == 0) atomicAdd(ws + 2, tot);
  }
  // per-class counters (counts < 2^24, exact in f32)
  if (threadIdx.x < 23)
    atomicAdd(ws + 3 + threadIdx.x, (float)cnt[threadIdx.x]);
}

// ---- kernel 3: finalize 27 outputs ---------------------------------------
// out: [loss, sent_loss, correct_dict(7), predict_dict(7), correct_total(7),
//       correct, count, 0, 0]
__global__ void finalize_kernel(const float* __restrict__ ws,
                                float* __restrict__ out, float invB) {
  if (blockIdx.x == 0 && threadIdx.x == 0) {
    float ldl = ws[0] * invB;            // labeled_doc_loss
    float udl = ws[1] * invB;            // unlabeled_doc_loss (weight 0)
    float cntv = ws[25];
    float sl = (cntv != 0.0f) ? ws[2] / cntv : 0.0f;
    out[0] = 1.0f * ldl + 0.0f * udl + 10.0f * sl;   // W1=1, W2=10
    out[1] = sl;
#pragma unroll
    for (int j = 0; j < NUM_CLASSES; ++j) {
      out[2 + j]  = ws[17 + j];   // correct_dict
      out[9 + j]  = ws[10 + j];   // predict_dict
      out[16 + j] = ws[3 + j];    // correct_total
    }
    out[23] = ws[24];             // correct
    out[24] = ws[25];             // count
    out[25] = 0.0f;               // p
    out[26] = 0.0f;               // r
  }
}

// ---- launcher ------------------------------------------------------------
extern "C" void kernel_launch(void* const* d_in, const int* in_sizes, int n_in,
                              void* d_out, int out_size, void* d_ws, size_t ws_size,
                              hipStream_t stream) {
  const float* ld = (const float*)d_in[0];   // labeled_doc  [B,1]
  const float* t1 = (const float*)d_in[1];   // target1      [B]
  const float* ls = (const float*)d_in[2];   // labeled_sent [N,7]
  const int*   t2 = (const int*)  d_in[3];   // target2      [N]
  const float* ud = (const float*)d_in[4];   // unlabeled_doc[B,1]
  const float* t3 = (const float*)d_in[5];   // target3      [B]
  float* out = (float*)d_out;
  float* ws  = (float*)d_ws;

  const int B  = in_sizes[0];       // 1048576
  const int N  = in_sizes[3];       // 2097152 sentence rows
  const int n4 = B / 4;

  zero_ws_kernel<<<1, 32, 0, stream>>>(ws);

  int docBlocks = (n4 + 255) / 256;
  if (docBlocks > 2048) docBlocks = 2048;
  if (docBlocks < 1) docBlocks = 1;
  doc_mse_kernel<<<docBlocks, 256, 0, stream>>>(ld, t1, ud, t3, n4, B, ws);

  int sentBlocks = (N + 255) / 256;
  if (sentBlocks > 4096) sentBlocks = 4096;
  if (sentBlocks < 1) sentBlocks = 1;
  sent_loss_kernel<<<sentBlocks, 256, 0, stream>>>(ls, t2, N, ws);

  finalize_kernel<<<1, 32, 0, stream>>>(ws, out, 1.0f / (float)B);
}